// Feat3dNet_39307540693641
// MI455X (gfx1250) — compile-verified
//
#include <hip/hip_runtime.h>
#include <hip/hip_bf16.h>
#include <math.h>

typedef __attribute__((ext_vector_type(2))) float v2f;
typedef __attribute__((ext_vector_type(8))) float v8f;

// Only check the builtin on the DEVICE pass; host (x86) pass has no AMDGPU builtins.
#if defined(__HIP_DEVICE_COMPILE__)
#if !__has_builtin(__builtin_amdgcn_wmma_f32_16x16x4_f32)
#error "DEVICE: wmma_f32_16x16x4_f32 builtin not available on this toolchain"
#endif
#endif

#define B_      8
#define N_      8192
#define M_      512
#define S_      64
#define RAD2    4.0f

// d_out float offsets (outputs concatenated in return order)
#define OUT_NEWXYZ 0          // 8*512*3  = 12288
#define OUT_IDX    12288      // 8*512*64 = 262144
#define OUT_ATT    274432     // 8*512    = 4096
#define OUT_ORI    278528     // 8*512    = 4096

// d_ws float offsets: k-pair-packed weights {W[k][n],W[k+1][n]} adjacent
#define WSP_W2  0             // 32 pairs * 128 cols * 2 = 8192 floats
#define WSP_W3  8192          // 64 pairs * 256 cols * 2 = 32768 floats

__device__ __forceinline__ v8f wmma_f32(v2f a, v2f b, v8f c) {
#if defined(__HIP_DEVICE_COMPILE__)
    // (neg_a, A, neg_b, B, c_mod, C, reuse_a, reuse_b)
    return __builtin_amdgcn_wmma_f32_16x16x4_f32(false, a, false, b, (short)0, c,
                                                 false, false);
#else
    return c;
#endif
}

// ---------------------------------------------------------------------------
// Kernel 0: pack W2 (64x128) and W3 (128x256) into k-pair interleaved layout
// so a WMMA B-fragment is a single 8-byte load.
// ---------------------------------------------------------------------------
__global__ __launch_bounds__(256) void pack_weights_kernel(
    const float* __restrict__ W2, const float* __restrict__ W3,
    float* __restrict__ ws)
{
    const int t = blockIdx.x * 256 + threadIdx.x;
    if (t < 4096) {                       // W2 pairs: kp in [0,32), col in [0,128)
        const int kp = t >> 7, col = t & 127;
        ws[WSP_W2 + 2 * t + 0] = W2[(2 * kp + 0) * 128 + col];
        ws[WSP_W2 + 2 * t + 1] = W2[(2 * kp + 1) * 128 + col];
    } else if (t < 4096 + 16384) {        // W3 pairs: kp in [0,64), col in [0,256)
        const int p = t - 4096;
        const int kp = p >> 8, col = p & 255;
        ws[WSP_W3 + 2 * p + 0] = W3[(2 * kp + 0) * 256 + col];
        ws[WSP_W3 + 2 * p + 1] = W3[(2 * kp + 1) * 256 + col];
    }
}

// ---------------------------------------------------------------------------
// Kernel 1: ball query (TF semantics: first nsample points by index within
// radius; pad with first neighbor). One wave32 per cluster center.
// Writes new_xyz and idx (as float) directly into d_out.
// ---------------------------------------------------------------------------
__global__ __launch_bounds__(256) void ball_query_kernel(
    const float* __restrict__ xyz, float* __restrict__ out)
{
    __shared__ int sIdx[8][S_];
    const int wave = threadIdx.x >> 5;
    const int lane = threadIdx.x & 31;
    const int c = blockIdx.x * 8 + wave;   // 0..4095
    const int b = c >> 9;
    const int m = c & 511;
    const float* pb = xyz + (size_t)b * N_ * 3;
    const float cx = pb[m * 3 + 0];
    const float cy = pb[m * 3 + 1];
    const float cz = pb[m * 3 + 2];

    int count = 0;
    int first = 0;
    for (int i0 = 0; i0 < N_; i0 += 32) {
        const int i = i0 + lane;
        const float dx = pb[i * 3 + 0] - cx;
        const float dy = pb[i * 3 + 1] - cy;
        const float dz = pb[i * 3 + 2] - cz;
        const bool valid = (dx * dx + dy * dy + dz * dz) < RAD2;
        const unsigned mask = __builtin_amdgcn_ballot_w32(valid);
        if (count == 0 && mask != 0u) first = i0 + __builtin_ctz(mask);
        if (valid) {
            const int pos = count + __popc(mask & ((1u << lane) - 1u));
            if (pos < S_) sIdx[wave][pos] = i;
        }
        count += __popc(mask);           // uniform across wave
        if (count >= S_) break;          // uniform branch
    }
    if (count > S_) count = S_;
    for (int s = count + lane; s < S_; s += 32) sIdx[wave][s] = first;
    __syncthreads();

    if (lane == 0) {
        out[OUT_NEWXYZ + c * 3 + 0] = cx;
        out[OUT_NEWXYZ + c * 3 + 1] = cy;
        out[OUT_NEWXYZ + c * 3 + 2] = cz;
    }
    for (int s = lane; s < S_; s += 32)
        out[OUT_IDX + c * S_ + s] = (float)sIdx[wave][s];
}

// ---------------------------------------------------------------------------
// Kernel 2: fused MLP 3->64->128->256 (WMMA f32 16x16x4) + max-pool + head.
// One block (256 threads / 8 waves) per cluster; intermediates in LDS only.
// k-outer loops with 4 accumulators: B fragment loaded once per k-step.
// ---------------------------------------------------------------------------
__global__ __launch_bounds__(256) void feat_mlp_kernel(
    const float* __restrict__ xyz,
    const float* __restrict__ W1, const float* __restrict__ b1,
    const float* __restrict__ b2, const float* __restrict__ b3,
    const float* __restrict__ W4, const float* __restrict__ b4,
    const float* __restrict__ W5, const float* __restrict__ b5,
    const float* __restrict__ Wa, const float* __restrict__ ba,
    const float* __restrict__ Wo, const float* __restrict__ bo,
    const float* __restrict__ ws,        // packed W2/W3
    float* __restrict__ out)
{
    __shared__ float sG[S_ * 4];        //  1 KB : g, K padded 3->4
    __shared__ float sW1[4 * 64];       //  1 KB : W1 zero-padded to 4x64
    __shared__ float sH1[S_ * 64];      // 16 KB
    __shared__ float sH2[S_ * 128];     // 32 KB
    __shared__ float sPool[256];
    __shared__ float sH4[128];
    __shared__ float sH5[64];

    const int c = blockIdx.x;           // cluster 0..4095
    const int b = c >> 9;
    const int m = c & 511;
    const int t = threadIdx.x;
    const int wave = t >> 5;
    const int lane = t & 31;
    const int ln = lane & 15;           // N index within tile
    const int kh = lane >> 4;           // lane-half selects K group
    const float* pb = xyz + (size_t)b * N_ * 3;
    const float cx = pb[m * 3 + 0];
    const float cy = pb[m * 3 + 1];
    const float cz = pb[m * 3 + 2];
    const v2f* W2p = (const v2f*)(ws + WSP_W2);
    const v2f* W3p = (const v2f*)(ws + WSP_W3);

    // Phase 0: gather grouped coords via idx already in d_out; stage W1 in LDS
    if (t < S_) {
        const int j = (int)out[OUT_IDX + c * S_ + t];
        sG[t * 4 + 0] = (pb[j * 3 + 0] - cx) * 0.5f;
        sG[t * 4 + 1] = (pb[j * 3 + 1] - cy) * 0.5f;
        sG[t * 4 + 2] = (pb[j * 3 + 2] - cz) * 0.5f;
        sG[t * 4 + 3] = 0.0f;
    }
    {
        const int wrow = t >> 6, wcol = t & 63;       // 256 entries of 4x64
        sW1[t] = (wrow < 3) ? W1[wrow * 64 + wcol] : 0.0f;
    }
    __syncthreads();

    // Phase 1: H1 = relu(G @ W1 + b1)   (64x4 @ 4x64), one wmma per tile
    for (int ti = 0; ti < 2; ++ti) {
        const int tile = wave * 2 + ti;        // 16 tiles (4x4)
        const int tm = tile >> 2, tn = tile & 3;
        const int row = tm * 16 + ln;
        const int col = tn * 16 + ln;
        v2f a, bb;
        a.x = sG[row * 4 + 2 * kh + 0];
        a.y = sG[row * 4 + 2 * kh + 1];
        bb.x = sW1[(2 * kh + 0) * 64 + col];
        bb.y = sW1[(2 * kh + 1) * 64 + col];
        v8f acc = {};
        acc = wmma_f32(a, bb, acc);
        const float bias = b1[col];
        #pragma unroll
        for (int r = 0; r < 8; ++r) {
            const int mm = tm * 16 + kh * 8 + r;
            sH1[mm * 64 + col] = fmaxf(acc[r] + bias, 0.0f);
        }
    }
    __syncthreads();

    // Phase 2: H2 = relu(H1 @ W2 + b2)  (64x64 @ 64x128)
    // Each wave owns one 16-col n-tile (8 tiles = 128 cols); k outer, 4 m-accs.
    {
        const int col = wave * 16 + ln;
        v8f acc0 = {}, acc1 = {}, acc2 = {}, acc3 = {};
        for (int kk2 = kh; kk2 < 32; kk2 += 2) {   // k-pair index: k = 2*kk2
            const v2f bb = W2p[kk2 * 128 + col];
            v2f a;
            a.x = sH1[(0 * 16 + ln) * 64 + 2 * kk2 + 0];
            a.y = sH1[(0 * 16 + ln) * 64 + 2 * kk2 + 1];
            acc0 = wmma_f32(a, bb, acc0);
            a.x = sH1[(1 * 16 + ln) * 64 + 2 * kk2 + 0];
            a.y = sH1[(1 * 16 + ln) * 64 + 2 * kk2 + 1];
            acc1 = wmma_f32(a, bb, acc1);
            a.x = sH1[(2 * 16 + ln) * 64 + 2 * kk2 + 0];
            a.y = sH1[(2 * 16 + ln) * 64 + 2 * kk2 + 1];
            acc2 = wmma_f32(a, bb, acc2);
            a.x = sH1[(3 * 16 + ln) * 64 + 2 * kk2 + 0];
            a.y = sH1[(3 * 16 + ln) * 64 + 2 * kk2 + 1];
            acc3 = wmma_f32(a, bb, acc3);
        }
        const float bias = b2[col];
        #pragma unroll
        for (int r = 0; r < 8; ++r) {
            const int mr = kh * 8 + r;
            sH2[(0 * 16 + mr) * 128 + col] = fmaxf(acc0[r] + bias, 0.0f);
            sH2[(1 * 16 + mr) * 128 + col] = fmaxf(acc1[r] + bias, 0.0f);
            sH2[(2 * 16 + mr) * 128 + col] = fmaxf(acc2[r] + bias, 0.0f);
            sH2[(3 * 16 + mr) * 128 + col] = fmaxf(acc3[r] + bias, 0.0f);
        }
    }
    __syncthreads();

    // Phase 3: H3 = relu(H2 @ W3 + b3) fused with column-max pool.
    // Each wave owns 2 n-tiles (16 tiles = 256 cols); k outer, 4 m-accs.
    for (int ti = 0; ti < 2; ++ti) {
        const int tn = wave * 2 + ti;
        const int col = tn * 16 + ln;
        v8f acc0 = {}, acc1 = {}, acc2 = {}, acc3 = {};
        for (int kk2 = kh; kk2 < 64; kk2 += 2) {   // k-pair index: k = 2*kk2
            const v2f bb = W3p[kk2 * 256 + col];
            v2f a;
            a.x = sH2[(0 * 16 + ln) * 128 + 2 * kk2 + 0];
            a.y = sH2[(0 * 16 + ln) * 128 + 2 * kk2 + 1];
            acc0 = wmma_f32(a, bb, acc0);
            a.x = sH2[(1 * 16 + ln) * 128 + 2 * kk2 + 0];
            a.y = sH2[(1 * 16 + ln) * 128 + 2 * kk2 + 1];
            acc1 = wmma_f32(a, bb, acc1);
            a.x = sH2[(2 * 16 + ln) * 128 + 2 * kk2 + 0];
            a.y = sH2[(2 * 16 + ln) * 128 + 2 * kk2 + 1];
            acc2 = wmma_f32(a, bb, acc2);
            a.x = sH2[(3 * 16 + ln) * 128 + 2 * kk2 + 0];
            a.y = sH2[(3 * 16 + ln) * 128 + 2 * kk2 + 1];
            acc3 = wmma_f32(a, bb, acc3);
        }
        const float bias = b3[col];
        float colmax = 0.0f;                       // relu output >= 0
        #pragma unroll
        for (int r = 0; r < 8; ++r) {
            colmax = fmaxf(colmax, acc0[r] + bias);
            colmax = fmaxf(colmax, acc1[r] + bias);
            colmax = fmaxf(colmax, acc2[r] + bias);
            colmax = fmaxf(colmax, acc3[r] + bias);
        }
        colmax = fmaxf(colmax, 0.0f);
        // lane l holds rows 8*kh..8*kh+7 of column ln; combine halves
        colmax = fmaxf(colmax, __shfl_xor(colmax, 16, 32));
        if (kh == 0) sPool[tn * 16 + ln] = colmax;
    }
    __syncthreads();

    // Phase 4: per-cluster head (matvecs; M=1 so scalar is right)
    if (t < 128) {
        float acc = b4[t];
        for (int k = 0; k < 256; ++k) acc += sPool[k] * W4[k * 128 + t];
        sH4[t] = acc;
    }
    __syncthreads();
    if (t < 64) {
        float acc = b5[t];
        for (int k = 0; k < 128; ++k) acc += sH4[k] * W5[k * 64 + t];
        sH5[t] = acc;
    }
    __syncthreads();
    if (t == 0) {
        float sa = ba[0], ox = bo[0], oy = bo[1];
        for (int k = 0; k < 64; ++k) {
            const float h = sH5[k];
            sa += h * Wa[k];
            ox += h * Wo[k * 2 + 0];
            oy += h * Wo[k * 2 + 1];
        }
        // stable softplus
        const float att = (sa > 0.0f) ? sa + log1pf(expf(-sa)) : log1pf(expf(sa));
        out[OUT_ATT + c] = att;
        const float rn = rsqrtf(fmaxf(ox * ox + oy * oy, 1e-8f));
        out[OUT_ORI + c] = atan2f(oy * rn, ox * rn);
    }
}

extern "C" void kernel_launch(void* const* d_in, const int* in_sizes, int n_in,
                              void* d_out, int out_size, void* d_ws, size_t ws_size,
                              hipStream_t stream) {
    (void)in_sizes; (void)n_in; (void)ws_size; (void)out_size;
    const float* xyz = (const float*)d_in[0];
    const float* W1 = (const float*)d_in[1];  const float* b1 = (const float*)d_in[2];
    const float* W2 = (const float*)d_in[3];  const float* b2 = (const float*)d_in[4];
    const float* W3 = (const float*)d_in[5];  const float* b3 = (const float*)d_in[6];
    const float* W4 = (const float*)d_in[7];  const float* b4 = (const float*)d_in[8];
    const float* W5 = (const float*)d_in[9];  const float* b5 = (const float*)d_in[10];
    const float* Wa = (const float*)d_in[11]; const float* ba = (const float*)d_in[12];
    const float* Wo = (const float*)d_in[13]; const float* bo = (const float*)d_in[14];
    float* out = (float*)d_out;
    float* ws  = (float*)d_ws;

    // Pack W2/W3 into k-pair layout (one b64 per B fragment)
    pack_weights_kernel<<<80, 256, 0, stream>>>(W2, W3, ws);
    // 4096 clusters: 8 waves (centers) per block -> 512 blocks
    ball_query_kernel<<<512, 256, 0, stream>>>(xyz, out);
    // one block per cluster, fused MLP + pool + head
    feat_mlp_kernel<<<4096, 256, 0, stream>>>(xyz, W1, b1, b2, b3,
                                              W4, b4, W5, b5, Wa, ba, Wo, bo,
                                              ws, out);
}